// HyperTurbulentGenerator_8237747273957
// MI455X (gfx1250) — compile-verified
//
#include <hip/hip_runtime.h>
#include <hip/hip_bf16.h>

// Problem constants (match reference)
#define BB 64
#define DD 512
#define NBASES 8
#define SS 512
#define D2 (DD * DD)

typedef __attribute__((ext_vector_type(2))) float v2f;
typedef __attribute__((ext_vector_type(8))) float v8f;
typedef __attribute__((ext_vector_type(4))) int   v4i;

// GEMM tiling
#define BM 128           // workgroup rows (8 waves x 16)
#define BN 64            // workgroup cols (4 x 16 per wave)
#define KC 32            // K chunk staged in LDS
#define LDP 36           // padded LDS leading dim (floats): 16B-aligned rows,
                         // stride 36 mod 64 -> conflict-free fragment reads

// ---- CDNA5 async global->LDS staging ---------------------------------------
// Helpers are declared for BOTH host and device passes; the device-only
// builtins are selected inside, so the host pass still parses cleanly.
typedef __attribute__((address_space(1))) v4i gv4i;   // global b128
typedef __attribute__((address_space(3))) v4i lv4i;   // LDS b128

#if __has_builtin(__builtin_amdgcn_global_load_async_to_lds_b128) && \
    __has_builtin(__builtin_amdgcn_s_wait_asynccnt)
__device__ __forceinline__ void async_g2l_b128(const float* gp, float* lp) {
    __builtin_amdgcn_global_load_async_to_lds_b128((gv4i*)(gp), (lv4i*)(lp), 0, 0);
}
__device__ __forceinline__ void wait_async_all() {
    __builtin_amdgcn_s_wait_asynccnt(0);
}
#else
__device__ __forceinline__ void async_g2l_b128(const float* gp, float* lp) {
    *(float4*)lp = *(const float4*)gp;    // synchronous fallback
}
__device__ __forceinline__ void wait_async_all() {}
#endif

// ---------------------------------------------------------------------------
// Batched WMMA GEMM:  C = scale * op(A) @ op(B)  (+ I if addI)
// op(A): MxK, op(B): KxN, C: MxN.  N multiple of 64, K multiple of 32.
// Block = 256 threads = 8 waves; tile 128(M) x 64(N).
// Double-buffered LDS panels; async loads issued a chunk ahead of the WMMAs.
// ---------------------------------------------------------------------------
template <bool TA, bool TB>
__global__ __launch_bounds__(256)
void gemm_wmma_kernel(const float* __restrict__ A,
                      const float* __restrict__ B,
                      float* __restrict__ C,
                      int M, int N, int K,
                      long long strideA, long long strideB, long long strideC,
                      int lda, int ldb, int ldc,
                      float scale, int addI)
{
    __shared__ float As[2][BM * LDP];   // As[buf][m*LDP + k] (operand layout)
    __shared__ float Bs[2][BN * LDP];   // Bs[buf][n*LDP + k] (operand layout)

    constexpr bool ASYNC_A = !TA;       // NN-A: contiguous b128 -> async to LDS
    constexpr bool ASYNC_B = TB;        // NT-B: contiguous b128 -> async to LDS

    const int batch = blockIdx.z;
    A += (long long)batch * strideA;
    B += (long long)batch * strideB;
    C += (long long)batch * strideC;

    const int tid  = threadIdx.x;
    const int lane = tid & 31;
    const int wave = tid >> 5;                 // 0..7
    const int rowBase0 = blockIdx.y * BM;
    const int colBase  = blockIdx.x * BN;
    const int hi = (lane >= 16) ? 1 : 0;       // K-half of the fragment
    const int r  = lane & 15;                  // row (A) / col (B,C) in tile

    // staging index precomputation
    const int mqA = (tid & 31) * 4;            // TA-A: m quad
    const int kkA = tid >> 5;                  // TA-A: k
    const int kqA = (tid & 7) * 4;             // NN-A: k quad
    const int m0A = tid >> 3;                  // NN-A: m base
    const int kqB = (tid & 7) * 4;             // TB-B: k quad
    const int n0B = tid >> 3;                  // TB-B: n base
    const int nqB = (tid & 15) * 4;            // NB-B: n quad
    const int kkB = tid >> 4;                  // NB-B: k base

    float4 ha[4];                              // held A panel (manual paths)
    float4 hb[2];                              // held B panel (manual paths)

    // ---- issue stage: async ops go straight to LDS buffer nxt; manual paths load
    //      into registers (committed to LDS after the compute phase).
    auto issue_stage = [&](int k0c, int nxt) {
        if (TA) {
            // opA(m,k) = A[k*lda + m] : contiguous in m
#pragma unroll
            for (int p = 0; p < 4; ++p) {
                const int k = kkA + p * 8;
                ha[p] = *(const float4*)(A + (long long)(k0c + k) * lda
                                           + (rowBase0 + mqA));
            }
        } else {
            // opA(m,k) = A[m*lda + k] : contiguous in k
#pragma unroll
            for (int p = 0; p < 4; ++p) {
                const int m = m0A + p * 32;
                const int row = rowBase0 + m;
                if (ASYNC_A) {
                    if (row < M) {
                        async_g2l_b128(A + (long long)row * lda + (k0c + kqA),
                                       &As[nxt][m * LDP + kqA]);
                    } else {
                        As[nxt][m * LDP + kqA + 0] = 0.0f;
                        As[nxt][m * LDP + kqA + 1] = 0.0f;
                        As[nxt][m * LDP + kqA + 2] = 0.0f;
                        As[nxt][m * LDP + kqA + 3] = 0.0f;
                    }
                } else {
                    if (row < M)
                        ha[p] = *(const float4*)(A + (long long)row * lda + (k0c + kqA));
                    else
                        ha[p] = make_float4(0.0f, 0.0f, 0.0f, 0.0f);
                }
            }
        }
        if (TB) {
            // opB(k,n) = B[n*ldb + k] : contiguous in k
#pragma unroll
            for (int p = 0; p < 2; ++p) {
                const int n = n0B + p * 32;
                if (ASYNC_B) {
                    async_g2l_b128(B + (long long)(colBase + n) * ldb + (k0c + kqB),
                                   &Bs[nxt][n * LDP + kqB]);
                } else {
                    hb[p] = *(const float4*)(B + (long long)(colBase + n) * ldb
                                               + (k0c + kqB));
                }
            }
        } else {
            // opB(k,n) = B[k*ldb + n] : contiguous in n
#pragma unroll
            for (int p = 0; p < 2; ++p) {
                const int k = kkB + p * 16;
                hb[p] = *(const float4*)(B + (long long)(k0c + k) * ldb
                                           + (colBase + nqB));
            }
        }
    };

    // ---- commit stage: scatter/store held registers into LDS buffer nxt
    auto commit_stage = [&](int nxt) {
        if (TA) {
#pragma unroll
            for (int p = 0; p < 4; ++p) {
                const int k = kkA + p * 8;
                As[nxt][(mqA + 0) * LDP + k] = ha[p].x;
                As[nxt][(mqA + 1) * LDP + k] = ha[p].y;
                As[nxt][(mqA + 2) * LDP + k] = ha[p].z;
                As[nxt][(mqA + 3) * LDP + k] = ha[p].w;
            }
        } else if (!ASYNC_A) {
#pragma unroll
            for (int p = 0; p < 4; ++p) {
                const int m = m0A + p * 32;
                *(float4*)&As[nxt][m * LDP + kqA] = ha[p];
            }
        }
        if (TB) {
            if (!ASYNC_B) {
#pragma unroll
                for (int p = 0; p < 2; ++p) {
                    const int n = n0B + p * 32;
                    *(float4*)&Bs[nxt][n * LDP + kqB] = hb[p];
                }
            }
        } else {
#pragma unroll
            for (int p = 0; p < 2; ++p) {
                const int k = kkB + p * 16;
                Bs[nxt][(nqB + 0) * LDP + k] = hb[p].x;
                Bs[nxt][(nqB + 1) * LDP + k] = hb[p].y;
                Bs[nxt][(nqB + 2) * LDP + k] = hb[p].z;
                Bs[nxt][(nqB + 3) * LDP + k] = hb[p].w;
            }
        }
    };

    v8f acc0 = {}, acc1 = {}, acc2 = {}, acc3 = {};
    const int mrow = wave * 16 + r;

    // ---- prologue: fill buffer 0 ----
    issue_stage(0, 0);
    commit_stage(0);
    wait_async_all();
    __syncthreads();

    int cur = 0;
    for (int k0 = 0; k0 < K; k0 += KC) {
        const int nxt = cur ^ 1;
        const bool havenext = (k0 + KC < K);

        // issue next chunk's loads before computing the current one
        if (havenext) issue_stage(k0 + KC, nxt);

        // ---- consume chunk: 8 x (5 ds_load_b64 + 4 wmma) per wave ---------
        const float* Asc = &As[cur][0];
        const float* Bsc = &Bs[cur][0];
#pragma unroll
        for (int kk4 = 0; kk4 < KC; kk4 += 4) {
            const int kfi = kk4 + 2 * hi;
            const v2f af = *(const v2f*)&Asc[mrow * LDP + kfi];
            const v2f b0 = *(const v2f*)&Bsc[(r +  0) * LDP + kfi];
            const v2f b1 = *(const v2f*)&Bsc[(r + 16) * LDP + kfi];
            const v2f b2 = *(const v2f*)&Bsc[(r + 32) * LDP + kfi];
            const v2f b3 = *(const v2f*)&Bsc[(r + 48) * LDP + kfi];
            acc0 = __builtin_amdgcn_wmma_f32_16x16x4_f32(false, af, false, b0,
                                                         (short)0, acc0, false, false);
            acc1 = __builtin_amdgcn_wmma_f32_16x16x4_f32(false, af, false, b1,
                                                         (short)0, acc1, false, false);
            acc2 = __builtin_amdgcn_wmma_f32_16x16x4_f32(false, af, false, b2,
                                                         (short)0, acc2, false, false);
            acc3 = __builtin_amdgcn_wmma_f32_16x16x4_f32(false, af, false, b3,
                                                         (short)0, acc3, false, false);
        }

        // park held registers into the next buffer, then synchronize
        if (havenext) commit_stage(nxt);
        wait_async_all();
        __syncthreads();
        cur = nxt;
    }

    // C/D layout: VGPR v, lanes 0-15 -> M=v, lanes 16-31 -> M=v+8, N=lane%16
    v8f accs[4] = {acc0, acc1, acc2, acc3};
#pragma unroll
    for (int nt = 0; nt < 4; ++nt) {
#pragma unroll
        for (int v = 0; v < 8; ++v) {
            const int row = rowBase0 + wave * 16 + v + 8 * hi;
            const int col = colBase + nt * 16 + r;
            if (row < M) {
                float val = accs[nt][v] * scale;
                if (addI && row == col) val += 1.0f;
                C[(long long)row * ldc + col] = val;
            }
        }
    }
}

// ---------------------------------------------------------------------------
// out[b] = sum_n coeffs[b][n] * bases[n]   (B x D x D)
// ---------------------------------------------------------------------------
__global__ void wsum_kernel(const float* __restrict__ coeffs,
                            const float* __restrict__ bases,
                            float* __restrict__ out)
{
    const int b = blockIdx.y;
    const long long idx = (long long)blockIdx.x * blockDim.x + threadIdx.x;
    float s = 0.0f;
#pragma unroll
    for (int n = 0; n < NBASES; ++n)
        s = fmaf(coeffs[b * NBASES + n], bases[(long long)n * D2 + idx], s);
    out[(long long)b * D2 + idx] = s;
}

// ---------------------------------------------------------------------------
// X[b] = (alpha[b]*(Ks - Ks^T) - beta[b]*RtR) * dt * inv2s    (in-place)
// ---------------------------------------------------------------------------
__global__ void build_a_kernel(const float* __restrict__ Ks,
                               float* __restrict__ X,
                               const float* __restrict__ alpha,
                               const float* __restrict__ beta,
                               const float* __restrict__ ts,
                               float inv2s)
{
    const int b = blockIdx.y;
    const long long idx = (long long)blockIdx.x * blockDim.x + threadIdx.x;
    const int i = (int)(idx >> 9);
    const int j = (int)(idx & (DD - 1));
    const float dt = ts[1] - ts[0];
    const float a  = alpha[b];
    const float be = beta[b];
    const float* K = Ks + (long long)b * D2;
    const float skew = K[(long long)i * DD + j] - K[(long long)j * DD + i];
    const long long off = (long long)b * D2 + idx;
    X[off] = (a * skew - be * X[off]) * dt * inv2s;
}

// ---------------------------------------------------------------------------
// P = I + X * invdeg    (Horner seed)
// ---------------------------------------------------------------------------
__global__ void initp_kernel(const float* __restrict__ X,
                             float* __restrict__ P, float invdeg)
{
    const int b = blockIdx.y;
    const long long idx = (long long)blockIdx.x * blockDim.x + threadIdx.x;
    const int i = (int)(idx >> 9);
    const int j = (int)(idx & (DD - 1));
    const long long off = (long long)b * D2 + idx;
    P[off] = X[off] * invdeg + ((i == j) ? 1.0f : 0.0f);
}

// ---------------------------------------------------------------------------
// out[b][0][:] = M[b] @ z0[b]
// ---------------------------------------------------------------------------
__global__ void matvec_kernel(const float* __restrict__ Mm,
                              const float* __restrict__ z0,
                              float* __restrict__ out)
{
    const int b = blockIdx.y;
    const int row = blockIdx.x * blockDim.x + threadIdx.x;
    const float* Mb = Mm + (long long)b * D2 + (long long)row * DD;
    const float* z  = z0 + (long long)b * DD;
    float s = 0.0f;
    for (int j = 0; j < DD; ++j) s = fmaf(Mb[j], z[j], s);
    out[(long long)b * SS * DD + row] = s;
}

// ---------------------------------------------------------------------------
extern "C" void kernel_launch(void* const* d_in, const int* in_sizes, int n_in,
                              void* d_out, int out_size, void* d_ws, size_t ws_size,
                              hipStream_t stream)
{
    const float* z0    = (const float*)d_in[0];
    const float* ts    = (const float*)d_in[1];
    const float* kco   = (const float*)d_in[2];
    const float* rco   = (const float*)d_in[3];
    const float* alpha = (const float*)d_in[4];
    const float* beta  = (const float*)d_in[5];
    const float* Kb    = (const float*)d_in[6];
    const float* Rb    = (const float*)d_in[7];
    float* out = (float*)d_out;

    float* ws  = (float*)d_ws;
    const size_t MAT = (size_t)BB * D2;
    float* ws0 = ws + 0 * MAT;              // Ksum  -> later: power ping
    float* ws1 = ws + 1 * MAT;              // Rsum  -> later: power pong
    float* ws2 = ws + 2 * MAT;              // R^T R -> scaled A (X)
    float* ws3 = ws + 3 * MAT;              // P ping
    float* ws4 = ws + 4 * MAT;              // P pong

    const dim3 eb(256);
    const dim3 eg(D2 / 256, BB);
    const dim3 gb(256);                     // 8 waves
    const dim3 gg(DD / BN, DD / BM, BB);    // full 512x512 batched GEMM

    // 1) K_sum, R_sum
    wsum_kernel<<<eg, eb, 0, stream>>>(kco, Kb, ws0);
    wsum_kernel<<<eg, eb, 0, stream>>>(rco, Rb, ws1);

    // 2) ws2 = R_sum^T @ R_sum
    gemm_wmma_kernel<true, false><<<gg, gb, 0, stream>>>(
        ws1, ws1, ws2, DD, DD, DD, (long long)D2, (long long)D2, (long long)D2,
        DD, DD, DD, 1.0f, 0);

    // 3) ws2 = (alpha*(Ks-Ks^T) - beta*RtR) * dt / 2^8   (expm scaling, s=8)
    build_a_kernel<<<eg, eb, 0, stream>>>(ws0, ws2, alpha, beta, ts, 1.0f / 256.0f);

    // 4) expm via Taylor(12) Horner + 8 squarings
    const int DEG = 12;
    initp_kernel<<<eg, eb, 0, stream>>>(ws2, ws3, 1.0f / (float)DEG);
    float* Pc = ws3;
    float* Po = ws4;
    for (int j = DEG - 1; j >= 1; --j) {
        gemm_wmma_kernel<false, false><<<gg, gb, 0, stream>>>(
            ws2, Pc, Po, DD, DD, DD, (long long)D2, (long long)D2, (long long)D2,
            DD, DD, DD, 1.0f / (float)j, 1);
        float* t = Pc; Pc = Po; Po = t;
    }
    for (int i = 0; i < 8; ++i) {
        gemm_wmma_kernel<false, false><<<gg, gb, 0, stream>>>(
            Pc, Pc, Po, DD, DD, DD, (long long)D2, (long long)D2, (long long)D2,
            DD, DD, DD, 1.0f, 0);
        float* t = Pc; Pc = Po; Po = t;
    }
    const float* Mmat = Pc;                 // M = expm(A*dt)

    // 5) row 0: z_1 = M @ z0
    matvec_kernel<<<dim3(DD / 256, BB), dim3(256), 0, stream>>>(Mmat, z0, out);

    // 6) log-doubling recurrence:
    //    rows[k..2k) = rows[0..k) @ (M^k)^T ;  M^{2k} = (M^k)^2
    const float* Mk = Mmat;
    float* pa = ws0;
    float* pb = ws1;
    for (int k = 1; k <= 256; k <<= 1) {
        dim3 gr(DD / BN, (k + BM - 1) / BM, BB);
        gemm_wmma_kernel<false, true><<<gr, gb, 0, stream>>>(
            out, Mk, out + (long long)k * DD,
            k, DD, DD,
            (long long)SS * DD, (long long)D2, (long long)SS * DD,
            DD, DD, DD, 1.0f, 0);
        if (k < 256) {
            gemm_wmma_kernel<false, false><<<gg, gb, 0, stream>>>(
                Mk, Mk, pa, DD, DD, DD, (long long)D2, (long long)D2, (long long)D2,
                DD, DD, DD, 1.0f, 0);
            Mk = pa;
            float* t = pa; pa = pb; pb = t;
        }
    }
    (void)in_sizes; (void)n_in; (void)out_size; (void)ws_size;
}